// DCNAlignNet_35665408426774
// MI455X (gfx1250) — compile-verified
//
#include <hip/hip_runtime.h>
#include <math.h>

#define HH 128
#define WW 128
#define HW 16384
#define NB 2
#define NFRAMES 5
#define CC 64
#define DGn 8
#define CGn 8
#define CENTER 2
#define CHUNK 16          // input channels staged per LDS round (conv)
#define COLS 66           // 64-pixel tile + 1 halo col each side
#define CELL 18           // padded per-(row,col) channel stride (bank spread)

typedef __attribute__((ext_vector_type(2))) float v2f;
typedef __attribute__((ext_vector_type(8))) float v8f;

// ---------------------------------------------------------------------------
// 3x3 SAME conv as implicit GEMM on the WMMA pipe (fp32 in / fp32 acc).
// Block = 4 waves = 64 contiguous pixels of one image row x 16 out-channels.
// Input staged in LDS in 16-channel chunks. Weights are pre-repacked to
// tap-major [o][t*Cin+ci], so each B fragment is one aligned b64 load; halo
// zeros are written at stage time and bias is folded into the accumulator:
// the inner loop is exactly ds_load_b64 + global_load_b64 + wmma.
// ---------------------------------------------------------------------------
__global__ __launch_bounds__(128)
void conv3x3_wmma(const float* __restrict__ x1, long x1Stride, int Cin1,
                  const float* __restrict__ x2, long x2Stride,
                  const float* __restrict__ wt, const float* __restrict__ bias,
                  float* __restrict__ out, long outStride,
                  int Cin, int Cout) {
    __shared__ float lds[3 * COLS * CELL];   // [(row*COLS+col)*CELL + ci]

    const int tid  = threadIdx.x;
    const int wave = tid >> 5;
    const int lane = tid & 31;
    const int b    = blockIdx.z;
    const int pixBase = blockIdx.x * 64;     // 64 pixels, all in one row
    const int row0 = pixBase >> 7;           // W = 128
    const int col0 = pixBase & (WW - 1);
    const int coBase = blockIdx.y << 4;

    const int m   = lane & 15;
    const int hi  = lane >> 4;
    const int cit = wave * 16 + m;           // column within the 64-px tile
    const int co  = coBase + m;              // always a valid (padded) row
    const int K = Cin * 9;

    const float* xb1  = x1 + (size_t)b * x1Stride;
    const float* xb2  = x2 + (size_t)b * x2Stride;
    const float* wrow = wt + (size_t)co * K;  // tap-major: [t*Cin + ci]

    const float bb = bias[co];
    v8f acc = {bb, bb, bb, bb, bb, bb, bb, bb};

    const int nChunks = Cin / CHUNK;
    for (int ch = 0; ch < nChunks; ++ch) {
        const int ciBase = ch * CHUNK;
        // whole chunk comes from one tensor (Cin1 is a multiple of CHUNK)
        const float* src = (ciBase < Cin1)
                               ? (xb1 + (size_t)ciBase * HW)
                               : (xb2 + (size_t)(ciBase - Cin1) * HW);
        __syncthreads();
        // cooperative stage: 16 ch x 3 rows x 66 cols, halo zeros folded in
        for (int e = tid; e < CHUNK * 3 * COLS; e += 128) {
            int ci  = e / (3 * COLS);
            int r   = e - ci * (3 * COLS);
            int row = r / COLS;
            int col = r - row * COLS;
            int y = row0 + row - 1;
            int x = col0 + col - 1;
            int yc = min(max(y, 0), HH - 1);
            int xc = min(max(x, 0), WW - 1);
            float ok = (((unsigned)y < (unsigned)HH) &
                        ((unsigned)x < (unsigned)WW)) ? 1.0f : 0.0f;
            lds[(row * COLS + col) * CELL + ci] =
                src[(size_t)ci * HW + yc * WW + xc] * ok;
        }
        __syncthreads();

#pragma unroll
        for (int t = 0; t < 9; ++t) {
            const int trow = t / 3;               // compile-time constants
            const int tcol = t % 3;
            const int cellBase = (trow * COLS + tcol + cit) * CELL;
            const float* wtap = wrow + t * Cin + ciBase;
#pragma unroll
            for (int k0 = 0; k0 < CHUNK; k0 += 4) {
                const int ca = k0 + 2 * hi;       // even -> 8B-aligned pair
                v2f av = *(const v2f*)&lds[cellBase + ca];
                v2f bv = *(const v2f*)&wtap[ca];  // adjacent pair: one b64
                acc = __builtin_amdgcn_wmma_f32_16x16x4_f32(
                          false, av, false, bv, (short)0, acc, false, false);
            }
        }
    }

    if (co < Cout) {
        float* dst = out + (size_t)b * outStride + (size_t)co * HW
                     + (size_t)pixBase + wave * 16 + 8 * hi;
#pragma unroll
        for (int v = 0; v < 8; ++v) dst[v] = acc[v];
    }
}

// ---------------------------------------------------------------------------
// Deform-conv einsum GEMM: out[co][pix] = sum_k W[co][k] * samp[k][pix] + b.
// Block = 16 pixels; its 4 waves cover all 64 output channels, so the A chunk
// (64 k-rows x 16 px, 4 KB) is staged once in LDS and shared 4x. Weight pairs
// are naturally adjacent -> b64 global loads; A pairs adjacent in padded LDS.
// ---------------------------------------------------------------------------
__global__ __launch_bounds__(128)
void gemm_wmma(const float* __restrict__ A, long aStride,
               const float* __restrict__ w, const float* __restrict__ bias,
               float* __restrict__ out, long outStride, int K) {
    __shared__ float ldsA[16 * COLS];        // [pix][66] (padded rows)

    const int tid  = threadIdx.x;
    const int wave = tid >> 5;
    const int lane = tid & 31;
    const int b    = blockIdx.z;
    const int pixBase = blockIdx.x << 4;
    const int m  = lane & 15;
    const int hi = lane >> 4;
    const int co = (wave << 4) + m;          // Cout = 64 exactly

    const float* ab   = A + (size_t)b * aStride + pixBase;
    const float* wrow = w + (size_t)co * K;

    const float bb = bias[co];
    v8f acc = {bb, bb, bb, bb, bb, bb, bb, bb};

    for (int kBase = 0; kBase < K; kBase += 64) {   // K = 576 -> 9 chunks
        __syncthreads();
#pragma unroll
        for (int j = 0; j < 8; ++j) {               // 1024 elems / 128 thr
            int e   = tid + j * 128;
            int kk  = e >> 4;
            int pxl = e & 15;
            ldsA[pxl * COLS + kk] = ab[(size_t)(kBase + kk) * HW + pxl];
        }
        __syncthreads();
#pragma unroll
        for (int k0 = 0; k0 < 64; k0 += 4) {
            const int ka = k0 + 2 * hi;
            v2f av = *(const v2f*)&ldsA[m * COLS + ka];
            v2f bv = *(const v2f*)&wrow[kBase + ka];
            acc = __builtin_amdgcn_wmma_f32_16x16x4_f32(
                      false, av, false, bv, (short)0, acc, false, false);
        }
    }

    float* dst = out + (size_t)b * outStride + (size_t)co * HW
                 + (size_t)pixBase + 8 * hi;
#pragma unroll
    for (int v = 0; v < 8; ++v) dst[v] = acc[v];
}

// ---------------------------------------------------------------------------
// Bilinear deformable sampling. Writes samp[(g*Cg+c)*9+kk][hw]; validity
// folded into corner weights, addresses clamped -> branch-free gathers.
// ---------------------------------------------------------------------------
__global__ __launch_bounds__(256)
void dcn_sample(const float* __restrict__ x, long xStride,
                const float* __restrict__ off, float* __restrict__ samp) {
    const int b   = blockIdx.z;
    const int idx = blockIdx.x * 256 + threadIdx.x;   // over DG*9*HW
    const int hw  = idx & (HW - 1);
    const int r   = idx >> 14;
    const int kk  = r % 9;
    const int g   = r / 9;
    const int yi  = hw >> 7;
    const int xi  = hw & (WW - 1);

    const float* ob = off + (size_t)b * (size_t)(144 * HW);
    const float oy = ob[(size_t)((g * 9 + kk) * 2 + 0) * HW + hw];
    const float ox = ob[(size_t)((g * 9 + kk) * 2 + 1) * HW + hw];
    const float py = (float)yi + (float)(kk / 3 - 1) + oy;
    const float px = (float)xi + (float)(kk % 3 - 1) + ox;

    const float y0f = floorf(py);
    const float x0f = floorf(px);
    const float wy = py - y0f;
    const float wx = px - x0f;
    const int y0 = (int)y0f;
    const int x0 = (int)x0f;
    const int y1 = y0 + 1;
    const int x1 = x0 + 1;

    const bool vy0 = (unsigned)y0 < (unsigned)HH;
    const bool vy1 = (unsigned)y1 < (unsigned)HH;
    const bool vx0 = (unsigned)x0 < (unsigned)WW;
    const bool vx1 = (unsigned)x1 < (unsigned)WW;

    const float w00 = (vy0 & vx0) ? (1.f - wy) * (1.f - wx) : 0.f;
    const float w01 = (vy0 & vx1) ? (1.f - wy) * wx        : 0.f;
    const float w10 = (vy1 & vx0) ? wy * (1.f - wx)        : 0.f;
    const float w11 = (vy1 & vx1) ? wy * wx                : 0.f;

    const int yc0 = min(max(y0, 0), HH - 1);
    const int yc1 = min(max(y1, 0), HH - 1);
    const int xc0 = min(max(x0, 0), WW - 1);
    const int xc1 = min(max(x1, 0), WW - 1);
    const int f00 = yc0 * WW + xc0;
    const int f01 = yc0 * WW + xc1;
    const int f10 = yc1 * WW + xc0;
    const int f11 = yc1 * WW + xc1;

    const float* xb = x + (size_t)b * xStride + (size_t)g * CGn * HW;
    float* sb = samp + (size_t)b * (size_t)(576 * HW)
                + ((size_t)g * CGn * 9 + (size_t)kk) * HW + hw;
#pragma unroll
    for (int c = 0; c < CGn; ++c) {
        const float* img = xb + (size_t)c * HW;
        float v = img[f00] * w00 + img[f01] * w01
                + img[f10] * w10 + img[f11] * w11;
        sb[(size_t)c * 9 * HW] = v;
    }
}

// ---------------------------------------------------------------------------
// Repack conv weights [o][ci][t] -> tap-major [o][t*Cin + ci], padding the
// out-channel dim to CoutPad rows (zeros). Also emits a padded bias vector.
// ---------------------------------------------------------------------------
__global__ __launch_bounds__(256)
void repack_w_kernel(const float* __restrict__ w, const float* __restrict__ bsrc,
                     float* __restrict__ wt, float* __restrict__ bdst,
                     int Cin, int Cout, int CoutPad) {
    int idx = blockIdx.x * 256 + threadIdx.x;     // CoutPad*Cin*9 elements
    int KN = Cin * 9;
    if (idx < CoutPad * KN) {
        int o  = idx / KN;
        int r  = idx - o * KN;
        int t  = r / Cin;
        int ci = r - t * Cin;
        wt[idx] = (o < Cout) ? w[(size_t)o * KN + ci * 9 + t] : 0.0f;
    }
    if (idx < CoutPad) bdst[idx] = (idx < Cout) ? bsrc[idx] : 0.0f;
}

__global__ __launch_bounds__(256)
void copy_center_kernel(const float* __restrict__ xc, float* __restrict__ out) {
    int idx = blockIdx.x * 256 + threadIdx.x;   // NB*3*HW = 98304
    int b = idx / (3 * HW);
    int rest = idx - b * (3 * HW);
    out[(size_t)b * (NFRAMES * 3 * HW) + (size_t)CENTER * 3 * HW + rest] = xc[idx];
}

extern "C" void kernel_launch(void* const* d_in, const int* in_sizes, int n_in,
                              void* d_out, int out_size, void* d_ws, size_t ws_size,
                              hipStream_t stream) {
    (void)in_sizes; (void)n_in; (void)out_size; (void)ws_size;

    const float* pf = (const float*)d_in[0];
    const float* xc = (const float*)d_in[1];
    const float* cr_w = (const float*)d_in[2];
    const float* cr_b = (const float*)d_in[3];
    const float* off_w[4] = {(const float*)d_in[4],  (const float*)d_in[8],
                             (const float*)d_in[12], (const float*)d_in[16]};
    const float* off_b[4] = {(const float*)d_in[5],  (const float*)d_in[9],
                             (const float*)d_in[13], (const float*)d_in[17]};
    const float* dc_w[4]  = {(const float*)d_in[6],  (const float*)d_in[10],
                             (const float*)d_in[14], (const float*)d_in[18]};
    const float* dc_b[4]  = {(const float*)d_in[7],  (const float*)d_in[11],
                             (const float*)d_in[15], (const float*)d_in[19]};
    const float* rec_w = (const float*)d_in[20];
    const float* rec_b = (const float*)d_in[21];
    float* outp = (float*)d_out;

    // Workspace (floats): feaA | feaB | off | samp | repacked weights/biases
    float* feaA = (float*)d_ws;
    float* feaB = feaA + (size_t)NB * CC * HW;
    float* offb = feaB + (size_t)NB * CC * HW;
    float* samp = offb + (size_t)NB * 144 * HW;
    float* wtCr = samp + (size_t)NB * 576 * HW;        // 64 * 1152
    float* bCr  = wtCr + (size_t)64 * 1152;            // 64
    float* wtOf[4]; float* bOf[4];
    {
        float* p = bCr + 64;
        for (int j = 0; j < 4; ++j) {
            wtOf[j] = p;            p += (size_t)144 * 576;
            bOf[j]  = p;            p += 144;
        }
        // rec: pad to 16 rows
        wtCr[0] = wtCr[0];          // no-op
    }
    float* wtRe = bOf[3] + 144;                        // 16 * 576
    float* bRe  = wtRe + (size_t)16 * 576;             // 16

    const long pfStride   = (long)NFRAMES * CC * HW;
    const long feaStride  = (long)CC * HW;
    const long offStride  = (long)144 * HW;
    const long sampStride = (long)576 * HW;

    const dim3 blk(128);
    const dim3 gridC64(HW / 64, CC / 16, NB);    // 64-out-channel convs
    const dim3 gridC144(HW / 64, 144 / 16, NB);  // offset convs
    const dim3 gridC3(HW / 64, 1, NB);           // rec conv (padded weights)
    const dim3 gridGemm(HW / 16, 1, NB);         // deform GEMMs (Cout=64)
    const dim3 gridSamp((DGn * 9 * HW) / 256, 1, NB);

    // One-time weight repacks (tap-major, bias copies, rec padded to 16 rows)
    repack_w_kernel<<<dim3((64 * 1152 + 255) / 256), dim3(256), 0, stream>>>(
        cr_w, cr_b, wtCr, bCr, 2 * CC, CC, CC);
    for (int j = 0; j < 4; ++j)
        repack_w_kernel<<<dim3((144 * 576 + 255) / 256), dim3(256), 0, stream>>>(
            off_w[j], off_b[j], wtOf[j], bOf[j], CC, 144, 144);
    repack_w_kernel<<<dim3((16 * 576 + 255) / 256), dim3(256), 0, stream>>>(
        rec_w, rec_b, wtRe, bRe, CC, 3, 16);

    copy_center_kernel<<<dim3((NB * 3 * HW) / 256), dim3(256), 0, stream>>>(xc, outp);

    for (int i = 0; i < NFRAMES; ++i) {
        if (i == CENTER) continue;
        const float* ref  = pf + (size_t)CENTER * CC * HW;
        const float* supp = pf + (size_t)i * CC * HW;

        // fea0 = conv3(cat(ref, supp))             -> feaA
        conv3x3_wmma<<<gridC64, blk, 0, stream>>>(ref, pfStride, CC, supp, pfStride,
                                                  wtCr, bCr, feaA, feaStride, 2 * CC, CC);
        // off1 = conv3(fea0)                        -> offb
        conv3x3_wmma<<<gridC144, blk, 0, stream>>>(feaA, feaStride, CC, feaA, feaStride,
                                                   wtOf[0], bOf[0], offb, offStride, CC, 144);
        // fea1 = dconv(fea0, off1)                  -> feaB
        dcn_sample<<<gridSamp, dim3(256), 0, stream>>>(feaA, feaStride, offb, samp);
        gemm_wmma<<<gridGemm, blk, 0, stream>>>(samp, sampStride, dc_w[0], dc_b[0],
                                                feaB, feaStride, 576);
        // off2 = conv3(fea1); fea2 = dconv(fea1)    -> feaA
        conv3x3_wmma<<<gridC144, blk, 0, stream>>>(feaB, feaStride, CC, feaB, feaStride,
                                                   wtOf[1], bOf[1], offb, offStride, CC, 144);
        dcn_sample<<<gridSamp, dim3(256), 0, stream>>>(feaB, feaStride, offb, samp);
        gemm_wmma<<<gridGemm, blk, 0, stream>>>(samp, sampStride, dc_w[1], dc_b[1],
                                                feaA, feaStride, 576);
        // off3 = conv3(fea2); fea3 = dconv(supp)    -> feaB
        conv3x3_wmma<<<gridC144, blk, 0, stream>>>(feaA, feaStride, CC, feaA, feaStride,
                                                   wtOf[2], bOf[2], offb, offStride, CC, 144);
        dcn_sample<<<gridSamp, dim3(256), 0, stream>>>(supp, pfStride, offb, samp);
        gemm_wmma<<<gridGemm, blk, 0, stream>>>(samp, sampStride, dc_w[2], dc_b[2],
                                                feaB, feaStride, 576);
        // off4 = conv3(fea3); aligned = dconv(fea3) -> feaA
        conv3x3_wmma<<<gridC144, blk, 0, stream>>>(feaB, feaStride, CC, feaB, feaStride,
                                                   wtOf[3], bOf[3], offb, offStride, CC, 144);
        dcn_sample<<<gridSamp, dim3(256), 0, stream>>>(feaB, feaStride, offb, samp);
        gemm_wmma<<<gridGemm, blk, 0, stream>>>(samp, sampStride, dc_w[3], dc_b[3],
                                                feaA, feaStride, 576);
        // out_i = conv3(aligned, rec_padded)        -> d_out slice
        conv3x3_wmma<<<gridC3, blk, 0, stream>>>(feaA, feaStride, CC, feaA, feaStride,
                                                 wtRe, bRe,
                                                 outp + (size_t)i * 3 * HW,
                                                 (long)NFRAMES * 3 * HW, CC, 3);
    }
}